// EdgeWeightOptimizer_69097433858677
// MI455X (gfx1250) — compile-verified
//
#include <hip/hip_runtime.h>
#include <hip/hip_bf16.h>

// ---------------------------------------------------------------------------
// Problem constants (match reference): C=4096 communities, D=256 feature dim.
// ---------------------------------------------------------------------------
constexpr int C = 4096;
constexpr int D = 256;
constexpr float EPS = 1e-8f;

typedef __attribute__((ext_vector_type(2))) float v2f;
typedef __attribute__((ext_vector_type(8))) float v8f;

// ---------------------------------------------------------------------------
// Kernel 1: row-normalize feats -> fn.  One wave (32 lanes) per row of 256.
// blockDim = 256 (8 waves/block), grid = C/8 = 512.
// ---------------------------------------------------------------------------
__global__ void normalize_rows_kernel(const float* __restrict__ feats,
                                      float* __restrict__ fn) {
    const int lane = threadIdx.x & 31;
    const int wid  = threadIdx.x >> 5;
    const int row  = blockIdx.x * 8 + wid;

    const float* src = feats + (size_t)row * D;
    float vals[8];
    float ss = 0.0f;
#pragma unroll
    for (int i = 0; i < 8; ++i) {
        vals[i] = src[lane + 32 * i];
        ss += vals[i] * vals[i];
    }
    // butterfly reduce so every lane holds the row sum-of-squares
#pragma unroll
    for (int off = 16; off; off >>= 1)
        ss += __shfl_xor(ss, off, 32);

    const float inv = 1.0f / fmaxf(sqrtf(ss), EPS);

    float* dst = fn + (size_t)row * D;
#pragma unroll
    for (int i = 0; i < 8; ++i)
        dst[lane + 32 * i] = vals[i] * inv;
}

// ---------------------------------------------------------------------------
// Kernel 2: sim = fn * fn^T via V_WMMA_F32_16X16X4_F32 (full fp32 matrix op).
// Workgroup tile: 128(M) x 128(N).  8 waves; wave w owns rows [16w,16w+16) of
// the tile and all 128 columns (8 accumulators of 16x16).
// K is chunked by 64 through LDS; LDS row stride padded to 68 floats so that
// the per-lane fragment reads (one row per lane, consecutive K) are
// bank-conflict-free.
// ---------------------------------------------------------------------------
#define KCHUNK 64
#define LDS_STRIDE 68  // 64 + 4 pad -> bank = (4*row + col) % 64, conflict-free

__global__ void __launch_bounds__(256)
cosine_sim_wmma_kernel(const float* __restrict__ fn,
                       float* __restrict__ sim) {
    __shared__ float Af[128 * LDS_STRIDE];
    __shared__ float Bf[128 * LDS_STRIDE];

    const int tileM = blockIdx.y * 128;
    const int tileN = blockIdx.x * 128;

    const int t    = threadIdx.x;
    const int lane = t & 31;
    const int wid  = t >> 5;
    const int m    = lane & 15;        // M (for A) / N (for B) within sub-tile
    const int kh   = (lane >> 4) * 2;  // lanes 0-15 -> K base 0, lanes 16-31 -> K base 2

    v8f acc[8] = {};

    // global->LDS copy coordinates: 16 float4 per row-of-64, 16 rows per pass
    const int ldRow = t >> 4;          // 0..15
    const int ldCol = (t & 15) * 4;    // 0..60

    for (int kc = 0; kc < D; kc += KCHUNK) {
#pragma unroll
        for (int r8 = 0; r8 < 8; ++r8) {
            const int row = r8 * 16 + ldRow;
            const float4 a4 = *(const float4*)(fn + (size_t)(tileM + row) * D + kc + ldCol);
            const float4 b4 = *(const float4*)(fn + (size_t)(tileN + row) * D + kc + ldCol);
            *(float4*)(&Af[row * LDS_STRIDE + ldCol]) = a4;
            *(float4*)(&Bf[row * LDS_STRIDE + ldCol]) = b4;
        }
        __syncthreads();

        const float* aBase = &Af[(wid * 16 + m) * LDS_STRIDE];
#pragma unroll
        for (int ks = 0; ks < KCHUNK / 4; ++ks) {
            const int k0 = ks * 4 + kh;
            v2f a;
            a.x = aBase[k0];
            a.y = aBase[k0 + 1];
#pragma unroll
            for (int n = 0; n < 8; ++n) {
                const float* bBase = &Bf[(n * 16 + m) * LDS_STRIDE];
                v2f b;
                b.x = bBase[k0];
                b.y = bBase[k0 + 1];
                acc[n] = __builtin_amdgcn_wmma_f32_16x16x4_f32(
                    /*neg_a=*/false, a, /*neg_b=*/false, b,
                    /*c_mod=*/(short)0, acc[n],
                    /*reuse_a=*/false, /*reuse_b=*/false);
            }
        }
        __syncthreads();
    }

    // C/D layout: VGPR r -> M = r (lanes 0-15) or r+8 (lanes 16-31); N = lane&15
    const int rowBase = tileM + wid * 16 + (lane >> 4) * 8;
    const int colBase = tileN + m;
#pragma unroll
    for (int n = 0; n < 8; ++n) {
#pragma unroll
        for (int r = 0; r < 8; ++r) {
            sim[(size_t)(rowBase + r) * C + colBase + n * 16] = acc[n][r];
        }
    }
}

// ---------------------------------------------------------------------------
// Kernel 3: per-edge loss terms, deterministic block partial sums.
// grid = 4096 blocks x 256 threads, grid-stride over E edges.
// Diagonal handling: reference zeroes sim diagonal, so s = 0 when u == v.
// ---------------------------------------------------------------------------
__global__ void edge_loss_kernel(const float* __restrict__ ew,
                                 const float* __restrict__ sim,
                                 const int* __restrict__ u,
                                 const int* __restrict__ v,
                                 const float* __restrict__ counts,
                                 float* __restrict__ partials,
                                 int E) {
    float acc = 0.0f;
    for (int i = blockIdx.x * blockDim.x + threadIdx.x; i < E;
         i += gridDim.x * blockDim.x) {
        const int uu = u[i];
        const int vv = v[i];
        const size_t idx = (size_t)uu * C + vv;
        const float w = ew[idx];
        const float s = (uu == vv) ? 0.0f : sim[idx];
        const float cc = counts[i];
        const float d1 = w - cc;
        const float d2 = w - s;
        acc += d1 * d1 + d2 * d2;
    }
    // wave reduce (wave32)
#pragma unroll
    for (int off = 16; off; off >>= 1)
        acc += __shfl_down(acc, off, 32);

    __shared__ float wsum[8];
    const int lane = threadIdx.x & 31;
    const int wid  = threadIdx.x >> 5;
    if (lane == 0) wsum[wid] = acc;
    __syncthreads();
    if (threadIdx.x == 0) {
        float tot = 0.0f;
        for (int i = 0; i < 8; ++i) tot += wsum[i];
        partials[blockIdx.x] = tot;
    }
}

// ---------------------------------------------------------------------------
// Kernel 4: fixed-order reduction of 4096 partials -> scalar output.
// ALPHA = BETA = 1 so both loss sums are already combined.
// ---------------------------------------------------------------------------
__global__ void final_reduce_kernel(const float* __restrict__ partials,
                                    float* __restrict__ out) {
    __shared__ float sh[256];
    float s = 0.0f;
    for (int i = threadIdx.x; i < 4096; i += 256) s += partials[i];
    sh[threadIdx.x] = s;
    __syncthreads();
    for (int off = 128; off; off >>= 1) {
        if (threadIdx.x < off) sh[threadIdx.x] += sh[threadIdx.x + off];
        __syncthreads();
    }
    if (threadIdx.x == 0) out[0] = sh[0];
}

// ---------------------------------------------------------------------------
// Launch: inputs per reference order:
//   d_in[0] edge_weights [C*C] f32
//   d_in[1] new_feats    [C*D] f32
//   d_in[2] u            [E]   i32
//   d_in[3] v            [E]   i32
//   d_in[4] counts       [E]   f32
// Workspace: fn (4 MB) | sim (64 MB) | partials (16 KB)
// ---------------------------------------------------------------------------
extern "C" void kernel_launch(void* const* d_in, const int* in_sizes, int n_in,
                              void* d_out, int out_size, void* d_ws, size_t ws_size,
                              hipStream_t stream) {
    const float* ew     = (const float*)d_in[0];
    const float* feats  = (const float*)d_in[1];
    const int*   u      = (const int*)d_in[2];
    const int*   v      = (const int*)d_in[3];
    const float* counts = (const float*)d_in[4];
    const int    E      = in_sizes[2];
    float* out = (float*)d_out;

    char* ws = (char*)d_ws;
    float* fn       = (float*)ws;                                   // C*D f32
    float* sim      = (float*)(ws + (size_t)C * D * sizeof(float)); // C*C f32
    float* partials = (float*)(ws + (size_t)C * D * sizeof(float)
                                  + (size_t)C * C * sizeof(float)); // 4096 f32

    normalize_rows_kernel<<<C / 8, 256, 0, stream>>>(feats, fn);

    dim3 grid(C / 128, C / 128);
    cosine_sim_wmma_kernel<<<grid, 256, 0, stream>>>(fn, sim);

    edge_loss_kernel<<<4096, 256, 0, stream>>>(ew, sim, u, v, counts, partials, E);

    final_reduce_kernel<<<1, 256, 0, stream>>>(partials, out);
}